// TileGymJSD_4183298146646
// MI455X (gfx1250) — compile-verified
//
#include <hip/hip_runtime.h>
#include <hip/hip_bf16.h>
#include <math.h>
#include <stdint.h>

#define T_TOKENS     2048
#define VOCAB        32000
#define N4           (VOCAB / 4)   // 8000 float4 per row
#define IGNORE_INDEX (-100)
#define BLOCK        512
#define NWAVES       (BLOCK / 32)

// ---------------- CDNA5 async global->LDS helpers ----------------
// builtin signature (verified by hipcc): (v4i AS1* gsrc, v4i AS3* ldst,
// imm offset, imm cpol). Each active lane copies 16 bytes global->LDS;
// completion tracked with ASYNCcnt (s_wait_asynccnt).
typedef int v4i __attribute__((__vector_size__(4 * sizeof(int))));
typedef __attribute__((address_space(1))) v4i* gv4i_ptr;
typedef __attribute__((address_space(3))) v4i* lv4i_ptr;

__device__ __forceinline__ void async_load16(const float* g, float* l) {
#if __has_builtin(__builtin_amdgcn_global_load_async_to_lds_b128)
  __builtin_amdgcn_global_load_async_to_lds_b128(
      (gv4i_ptr)(uintptr_t)g,
      (lv4i_ptr)(uint32_t)(uintptr_t)l,   // low 32 bits of generic LDS ptr = LDS offset
      0, 0);
#else
  uint32_t lds_off = (uint32_t)(uintptr_t)l;
  asm volatile("global_load_async_to_lds_b128 %0, %1, off"
               :: "v"(lds_off), "v"(g) : "memory");
#endif
}

__device__ __forceinline__ void wait_async0() {
#if __has_builtin(__builtin_amdgcn_s_wait_asynccnt)
  __builtin_amdgcn_s_wait_asynccnt(0);
#else
  asm volatile("s_wait_asynccnt 0" ::: "memory");
#endif
}

// ---------------- main per-row kernel ----------------
// One block per token row. Both logit rows (2 x 128 KB) are staged into the
// 320 KB WGP LDS via async-DMA; all subsequent passes read LDS only via
// ds_load_b128, so HBM traffic is the 1x minimum (524 MB -> ~22.5 us floor
// @ 23.3 TB/s) and LDS-pass instruction count is 4x lower than scalar.
__global__ __launch_bounds__(BLOCK, 1)
void jsd_row_kernel(const float* __restrict__ input,    // student logits [T,V]
                    const float* __restrict__ target,   // teacher logits [T,V]
                    const int*   __restrict__ labels,   // [T]
                    float*       __restrict__ row_loss, // [T] (ws)
                    int*         __restrict__ row_mask) // [T] (ws)
{
  extern __shared__ float lds[];
  float* ls_q = lds;          // student row, VOCAB floats (16B aligned)
  float* ls_p = lds + VOCAB;  // teacher row, VOCAB floats (offset 128000B, 16B aligned)

  __shared__ float red0[NWAVES];
  __shared__ float red1[NWAVES];
  __shared__ float bc0, bc1;

  const int r    = blockIdx.x;
  const int tid  = threadIdx.x;
  const int wave = tid >> 5;
  const int lane = tid & 31;

  const float* gq = input  + (size_t)r * VOCAB;
  const float* gp = target + (size_t)r * VOCAB;

  // ---- Phase 0: async DMA both rows into LDS (16 B / lane / op) ----
  for (int i = tid; i < N4; i += BLOCK) {
    async_load16(gq + 4 * i, ls_q + 4 * i);
    async_load16(gp + 4 * i, ls_p + 4 * i);
  }
  wait_async0();     // my wave's async writes to LDS are done
  __syncthreads();   // everyone's are

  const float4* q4 = (const float4*)ls_q;
  const float4* p4 = (const float4*)ls_p;

  // ---- Phase A: row maxima (ds_load_b128, no transcendentals) ----
  float mq = -INFINITY, mp = -INFINITY;
  for (int i = tid; i < N4; i += BLOCK) {
    float4 a = q4[i];
    float4 b = p4[i];
    mq = fmaxf(mq, fmaxf(fmaxf(a.x, a.y), fmaxf(a.z, a.w)));
    mp = fmaxf(mp, fmaxf(fmaxf(b.x, b.y), fmaxf(b.z, b.w)));
  }
  for (int off = 16; off; off >>= 1) {
    mq = fmaxf(mq, __shfl_xor(mq, off, 32));
    mp = fmaxf(mp, __shfl_xor(mp, off, 32));
  }
  if (lane == 0) { red0[wave] = mq; red1[wave] = mp; }
  __syncthreads();
  if (tid == 0) {
    float a = red0[0], b = red1[0];
    for (int w = 1; w < NWAVES; ++w) { a = fmaxf(a, red0[w]); b = fmaxf(b, red1[w]); }
    bc0 = a; bc1 = b;
  }
  __syncthreads();
  const float Mq = bc0, Mp = bc1;
  __syncthreads();  // red0/red1 reused below

  // ---- Phase B: sum of exp(x - max)  (2 exp / element total) ----
  float sq = 0.f, sp = 0.f;
  for (int i = tid; i < N4; i += BLOCK) {
    float4 a = q4[i];
    float4 b = p4[i];
    sq += __expf(a.x - Mq) + __expf(a.y - Mq) + __expf(a.z - Mq) + __expf(a.w - Mq);
    sp += __expf(b.x - Mp) + __expf(b.y - Mp) + __expf(b.z - Mp) + __expf(b.w - Mp);
  }
  for (int off = 16; off; off >>= 1) {
    sq += __shfl_xor(sq, off, 32);
    sp += __shfl_xor(sp, off, 32);
  }
  if (lane == 0) { red0[wave] = sq; red1[wave] = sp; }
  __syncthreads();
  if (tid == 0) {
    float a = 0.f, b = 0.f;
    for (int w = 0; w < NWAVES; ++w) { a += red0[w]; b += red1[w]; }
    bc0 = Mq + __logf(a);   // lse of student row
    bc1 = Mp + __logf(b);   // lse of teacher row
  }
  __syncthreads();
  const float lse_q = bc0, lse_p = bc1;
  __syncthreads();  // red0 reused below

  // ---- Phase C: JSD sum. With beta = 1/2:
  //   beta*p*(lp-lm) + (1-beta)*q*(lq-lm) = 0.5*[p*lp + q*lq - (p+q)*(log(p+q)+log(1/2))]
  // p,q in (0,1] so log(p+q) is safe; 2 exp + 1 log per element.
  const float LOG_HALF = -0.6931471805599453f;
  float acc = 0.f;
  for (int i = tid; i < N4; i += BLOCK) {
    float4 a = q4[i];
    float4 b = p4[i];
    {
      float lq = a.x - lse_q, lp = b.x - lse_p;
      float u = __expf(lp), v = __expf(lq), s = u + v;
      acc += u * lp + v * lq - s * (__logf(s) + LOG_HALF);
    }
    {
      float lq = a.y - lse_q, lp = b.y - lse_p;
      float u = __expf(lp), v = __expf(lq), s = u + v;
      acc += u * lp + v * lq - s * (__logf(s) + LOG_HALF);
    }
    {
      float lq = a.z - lse_q, lp = b.z - lse_p;
      float u = __expf(lp), v = __expf(lq), s = u + v;
      acc += u * lp + v * lq - s * (__logf(s) + LOG_HALF);
    }
    {
      float lq = a.w - lse_q, lp = b.w - lse_p;
      float u = __expf(lp), v = __expf(lq), s = u + v;
      acc += u * lp + v * lq - s * (__logf(s) + LOG_HALF);
    }
  }
  for (int off = 16; off; off >>= 1) acc += __shfl_xor(acc, off, 32);
  if (lane == 0) red0[wave] = acc;
  __syncthreads();
  if (tid == 0) {
    float s = 0.f;
    for (int w = 0; w < NWAVES; ++w) s += red0[w];
    s *= 0.5f;  // the factored-out beta = 1/2
    int msk = (labels[r] != IGNORE_INDEX) ? 1 : 0;
    row_loss[r] = msk ? s : 0.f;
    row_mask[r] = msk;
  }
}

// ---------------- finalize: sum rows / max(count,1) ----------------
__global__ __launch_bounds__(256, 1)
void jsd_finalize(const float* __restrict__ row_loss,
                  const int*   __restrict__ row_mask,
                  float*       __restrict__ out)
{
  __shared__ float sacc[8];
  __shared__ int   scnt[8];
  float acc = 0.f; int cnt = 0;
  for (int i = threadIdx.x; i < T_TOKENS; i += 256) {
    acc += row_loss[i];
    cnt += row_mask[i];
  }
  for (int off = 16; off; off >>= 1) {
    acc += __shfl_xor(acc, off, 32);
    cnt += __shfl_xor(cnt, off, 32);
  }
  int wave = threadIdx.x >> 5, lane = threadIdx.x & 31;
  if (lane == 0) { sacc[wave] = acc; scnt[wave] = cnt; }
  __syncthreads();
  if (threadIdx.x == 0) {
    float s = 0.f; int c = 0;
    for (int w = 0; w < 8; ++w) { s += sacc[w]; c += scnt[w]; }
    out[0] = s / (float)((c > 0) ? c : 1);
  }
}

extern "C" void kernel_launch(void* const* d_in, const int* in_sizes, int n_in,
                              void* d_out, int out_size, void* d_ws, size_t ws_size,
                              hipStream_t stream) {
  (void)in_sizes; (void)n_in; (void)out_size; (void)ws_size;
  const float* input  = (const float*)d_in[0];
  const float* target = (const float*)d_in[1];
  const int*   labels = (const int*)d_in[2];
  float* out      = (float*)d_out;
  float* row_loss = (float*)d_ws;
  int*   row_mask = (int*)((char*)d_ws + T_TOKENS * sizeof(float));

  size_t shmem = (size_t)2 * VOCAB * sizeof(float);  // 256 000 B < 320 KB WGP LDS
  jsd_row_kernel<<<T_TOKENS, BLOCK, shmem, stream>>>(input, target, labels,
                                                     row_loss, row_mask);
  jsd_finalize<<<1, 256, 0, stream>>>(row_loss, row_mask, out);
}